// Patch_Embed_Outer_Rotate_64433099374843
// MI455X (gfx1250) — compile-verified
//
#include <hip/hip_runtime.h>

// Patch rotate: transpose every 16x16 patch except the inner 5x5 patch block.
// Pure permutation, HBM-bandwidth bound (~308 MB of traffic -> ~13 us at 23.3 TB/s).
// Strategy: geometry carried by the grid (no integer division), one thread per
// 4x4 fp32 sub-block; b128 NT loads, register 4x4 transpose, b128 NT stores.

#define PSZ 16   // patch size
#define IMG 224  // image H == W
#define NPD 14   // patches per dimension (224/16)
#define NCH 3    // channels

// Native clang vector type (required by __builtin_nontemporal_*).
typedef __attribute__((ext_vector_type(4))) float v4f;

// Block: 224 threads = 7 wave32 = one full row of 14 patches (16 threads/patch).
// grid.x = hp (patch row, 0..13), grid.y = plane (b*3 + c).
__global__ __launch_bounds__(224) void patch_outer_rotate_kernel(
    const float* __restrict__ x, float* __restrict__ out) {
  const int tid = threadIdx.x;   // 0..223
  const int wp  = tid >> 4;      // patch column 0..13 (per 16-lane group)
  const int sub = tid & 15;      // 4x4 sub-block id within patch
  const int r   = sub >> 2;      // sub-block row 0..3
  const int c   = sub & 3;       // sub-block col 0..3

  const int hp = blockIdx.x;     // patch row (scalar)
  const size_t plane_off = (size_t)blockIdx.y * (IMG * IMG);  // scalar

  // Inner 5x5 patch block (indices 5..9 in both dims) is copied, rest transposed.
  // hp test is wave-uniform; wp test is uniform per 16-lane half.
  const bool do_transpose = !(wp >= 5 && wp <= 9 && hp >= 5 && hp <= 9);

  const float* src = x + plane_off + (hp * PSZ + r * 4) * IMG + (wp * PSZ + c * 4);

  // Load 4 rows x 4 cols of the sub-block as b128 (non-temporal: streamed once).
  v4f v0 = __builtin_nontemporal_load((const v4f*)(src + 0 * IMG));
  v4f v1 = __builtin_nontemporal_load((const v4f*)(src + 1 * IMG));
  v4f v2 = __builtin_nontemporal_load((const v4f*)(src + 2 * IMG));
  v4f v3 = __builtin_nontemporal_load((const v4f*)(src + 3 * IMG));

  if (do_transpose) {
    // 4x4 register transpose; sub-block (r,c) lands at (c,r) in the patch.
    v4f t0  = {v0.x, v1.x, v2.x, v3.x};
    v4f t1v = {v0.y, v1.y, v2.y, v3.y};
    v4f t2  = {v0.z, v1.z, v2.z, v3.z};
    v4f t3  = {v0.w, v1.w, v2.w, v3.w};
    float* dst = out + plane_off + (hp * PSZ + c * 4) * IMG + (wp * PSZ + r * 4);
    __builtin_nontemporal_store(t0,  (v4f*)(dst + 0 * IMG));
    __builtin_nontemporal_store(t1v, (v4f*)(dst + 1 * IMG));
    __builtin_nontemporal_store(t2,  (v4f*)(dst + 2 * IMG));
    __builtin_nontemporal_store(t3,  (v4f*)(dst + 3 * IMG));
  } else {
    float* dst = out + plane_off + (hp * PSZ + r * 4) * IMG + (wp * PSZ + c * 4);
    __builtin_nontemporal_store(v0, (v4f*)(dst + 0 * IMG));
    __builtin_nontemporal_store(v1, (v4f*)(dst + 1 * IMG));
    __builtin_nontemporal_store(v2, (v4f*)(dst + 2 * IMG));
    __builtin_nontemporal_store(v3, (v4f*)(dst + 3 * IMG));
  }
}

extern "C" void kernel_launch(void* const* d_in, const int* in_sizes, int n_in,
                              void* d_out, int out_size, void* d_ws, size_t ws_size,
                              hipStream_t stream) {
  const float* x = (const float*)d_in[0];
  float* out     = (float*)d_out;

  // in_sizes[0] = B * 3 * 224 * 224 -> derive B.
  const int B = in_sizes[0] / (NCH * IMG * IMG);

  dim3 block(NPD * PSZ);          // 224 threads = 7 wave32
  dim3 grid(NPD, B * NCH);        // (patch row, plane) -> 14 x 768 for B=256

  patch_outer_rotate_kernel<<<grid, block, 0, stream>>>(x, out);
}